// RGNN_26027501814527
// MI455X (gfx1250) — compile-verified
//
#include <hip/hip_runtime.h>
#include <math.h>

// ---------------- problem constants (match reference) ----------------
#define N_NODES  50000
#define N_EDGES  800000
#define N_REL    3
#define N_LAYERS 3
#define IN_FEAT  64
#define N_HID    128
#define N_GRAPHS 32

typedef __attribute__((ext_vector_type(2))) float v2f;
typedef __attribute__((ext_vector_type(8))) float v8f;
typedef __attribute__((ext_vector_type(4))) int   v4i;

// ---- CDNA5 async global->LDS copy (ASYNCcnt-tracked), with fallback ----
#if __has_builtin(__builtin_amdgcn_global_load_async_to_lds_b128) && \
    __has_builtin(__builtin_amdgcn_s_wait_asynccnt)
#define USE_ASYNC_LDS 1
#else
#define USE_ASYNC_LDS 0
#endif

#if USE_ASYNC_LDS
typedef __attribute__((address_space(1))) v4i gv4i;   // global int4
typedef __attribute__((address_space(3))) v4i lv4i;   // LDS int4
__device__ __forceinline__ void async_copy16(const float* gsrc, float* ldst)
{
    __builtin_amdgcn_global_load_async_to_lds_b128(
        (gv4i*)gsrc, (lv4i*)ldst, 0, 0);
}
__device__ __forceinline__ void async_wait_all()
{
    __builtin_amdgcn_s_wait_asynccnt(0);
}
#else
__device__ __forceinline__ void async_copy16(const float* gsrc, float* ldst)
{
    *reinterpret_cast<float4*>(ldst) = *reinterpret_cast<const float4*>(gsrc);
}
__device__ __forceinline__ void async_wait_all() {}
#endif

// =====================================================================
// Fused row-major GEMM:  C[M,128] = A[M,K] @ W[K,128] (+ bias), fp32 WMMA.
// Block = 256 threads (8 waves). Wave w owns column tile n0 = 16*w.
// W staged in LDS once per block (stride 136: 16B-aligned, conflict-free
// B-fragment reads). 32-row node strips, grid-strided. Two 16x16
// accumulators per wave, K-loop of V_WMMA_F32_16X16X4_F32.
// Staging uses GLOBAL_LOAD_ASYNC_TO_LDS_B128 when available.
// =====================================================================
template <int K>
__global__ void __launch_bounds__(256)
gemm_wmma_f32(const float* __restrict__ A,
              const float* __restrict__ W,
              const float* __restrict__ bias,
              float* __restrict__ C,
              int M)
{
    constexpr int N  = 128;
    constexpr int WS = 136;      // LDS stride for W rows (pad: +8)
    constexpr int AS = K + 4;    // LDS stride for A rows (pad: +4)
    extern __shared__ float smem[];
    float* sW = smem;            // [K][WS]
    float* sA = smem + K * WS;   // [32][AS]

    const int tid  = threadIdx.x;
    const int lane = tid & 31;
    const int wave = tid >> 5;       // 0..7 -> column tile
    const int n0   = wave * 16;
    const int lhi  = lane >> 4;      // 0/1 : selects K sub-pair
    const int llo  = lane & 15;      // M (A) / N (B,C) index within tile

    // ---- stage weights into LDS (async b128, coalesced) ----
    for (int i = tid; i < K * (N / 4); i += 256) {
        const int k  = i >> 5;       // N/4 == 32
        const int n4 = i & 31;
        async_copy16(W + (size_t)i * 4, sW + k * WS + n4 * 4);
    }

    const float bv = bias ? bias[n0 + llo] : 0.0f;
    const int nStrips = (M + 31) >> 5;

    for (int strip = blockIdx.x; strip < nStrips; strip += gridDim.x) {
        const int row0 = strip << 5;
        __syncthreads();             // previous strip's LDS reads done
        // ---- stage 32 x K strip of A (rows clamped; extras unstored) ----
        for (int i = tid; i < 32 * (K / 4); i += 256) {
            const int r  = i / (K / 4);
            const int c4 = i - r * (K / 4);
            int row = row0 + r; if (row >= M) row = M - 1;
            async_copy16(A + (size_t)row * K + c4 * 4, sA + r * AS + c4 * 4);
        }
        async_wait_all();            // our async writes landed in LDS
        __syncthreads();             // all waves' writes visible (incl. sW)

        v8f c0 = {};   // rows [row0   , row0+16)
        v8f c1 = {};   // rows [row0+16, row0+32)
#pragma unroll
        for (int k0 = 0; k0 < K; k0 += 4) {
            const int ka = k0 + 2 * lhi;
            // B 4x16 fragment: vgpr j, lane -> W[k0 + j + 2*(lane/16)][n0+llo]
            v2f b;
            b.x = sW[ka * WS + n0 + llo];
            b.y = sW[(ka + 1) * WS + n0 + llo];
            // A 16x4 fragments: vgpr j, lane -> A[llo][k0 + j + 2*(lane/16)]
            v2f a0, a1;
            a0.x = sA[llo * AS + ka];
            a0.y = sA[llo * AS + ka + 1];
            a1.x = sA[(16 + llo) * AS + ka];
            a1.y = sA[(16 + llo) * AS + ka + 1];
            c0 = __builtin_amdgcn_wmma_f32_16x16x4_f32(
                     false, a0, false, b, (short)0, c0, false, false);
            c1 = __builtin_amdgcn_wmma_f32_16x16x4_f32(
                     false, a1, false, b, (short)0, c1, false, false);
        }

        // ---- epilogue: C layout vgpr v, lane -> row (v + 8*(lane/16)) ----
#pragma unroll
        for (int v = 0; v < 8; ++v) {
            const int rIn = v + 8 * lhi;
            int row = row0 + rIn;
            if (row < M) C[(size_t)row * N + n0 + llo] = c0[v] + bv;
            row = row0 + 16 + rIn;
            if (row < M) C[(size_t)row * N + n0 + llo] = c1[v] + bv;
        }
    }
}

// =====================================================================
// elementwise / scatter helpers (all float4-wide: HBM-bound stage)
// =====================================================================
__global__ void fill_u32x4_kernel(uint4* __restrict__ p, unsigned v, int n4)
{
    const int i = blockIdx.x * blockDim.x + threadIdx.x;
    if (i < n4) p[i] = make_uint4(v, v, v, v);
}

__device__ __forceinline__ void atomic_max_f32(float* addr, float v)
{
    // sign-split trick -> non-returning global_atomic_max/min, works with
    // -inf initialization (empty segments stay 0xFF800000).
    if (v >= 0.0f) atomicMax(reinterpret_cast<int*>(addr), __float_as_int(v));
    else atomicMin(reinterpret_cast<unsigned*>(addr), __float_as_uint(v));
}

// one wave (32 lanes) per edge; each lane moves a float4 of the 128-feature row
__global__ void __launch_bounds__(256)
scatter_max_kernel(const float* __restrict__ m,
                   const int* __restrict__ src,
                   const int* __restrict__ dst,
                   float* __restrict__ agg,
                   int nE)
{
    const int idx = blockIdx.x * blockDim.x + threadIdx.x;
    const int e = idx >> 5;
    if (e >= nE) return;
    const int f = (idx & 31) << 2;
    const int s = src[e];
    const int d = dst[e];
    const float4 v =
        *reinterpret_cast<const float4*>(m + (size_t)s * N_HID + f);
    float* o = agg + (size_t)d * N_HID + f;
    atomic_max_f32(o + 0, v.x);
    atomic_max_f32(o + 1, v.y);
    atomic_max_f32(o + 2, v.z);
    atomic_max_f32(o + 3, v.w);
}

// out += agg, with PyG semantics: empty segment (-inf) contributes 0
__global__ void add_agg_kernel(float4* __restrict__ out,
                               const float4* __restrict__ agg, int n4)
{
    const int i = blockIdx.x * blockDim.x + threadIdx.x;
    if (i >= n4) return;
    const float4 a = agg[i];
    float4 o = out[i];
    o.x += (a.x == -INFINITY) ? 0.0f : a.x;
    o.y += (a.y == -INFINITY) ? 0.0f : a.y;
    o.z += (a.z == -INFINITY) ? 0.0f : a.z;
    o.w += (a.w == -INFINITY) ? 0.0f : a.w;
    out[i] = o;
}

__global__ void relu_copy_kernel(float4* __restrict__ dst,
                                 const float4* __restrict__ src, int n4)
{
    const int i = blockIdx.x * blockDim.x + threadIdx.x;
    if (i >= n4) return;
    const float4 s = src[i];
    dst[i] = make_float4(fmaxf(s.x, 0.0f), fmaxf(s.y, 0.0f),
                         fmaxf(s.z, 0.0f), fmaxf(s.w, 0.0f));
}

// segment-sum pooling: one wave per node, float4 per lane, atomicAdd
__global__ void __launch_bounds__(256)
pool_sum_kernel(const float* __restrict__ h,
                const int* __restrict__ batch,
                float* __restrict__ pooled, int M)
{
    const int idx = blockIdx.x * blockDim.x + threadIdx.x;
    const int node = idx >> 5;
    if (node >= M) return;
    const int f = (idx & 31) << 2;
    const int g = batch[node];
    const float4 v =
        *reinterpret_cast<const float4*>(h + (size_t)node * N_HID + f);
    float* o = pooled + (size_t)g * N_HID + f;
    atomicAdd(o + 0, v.x);
    atomicAdd(o + 1, v.y);
    atomicAdd(o + 2, v.z);
    atomicAdd(o + 3, v.w);
}

// tiny MLP head: z = relu(pooled @ W1 + b1) @ W2 + b2  -> [32]
__global__ void __launch_bounds__(128)
mlp_head_kernel(const float* __restrict__ pooled,
                const float* __restrict__ W1, const float* __restrict__ b1,
                const float* __restrict__ W2, const float* __restrict__ b2,
                float* __restrict__ out)
{
    __shared__ float red[128];
    const int j = threadIdx.x;          // hidden index 0..127
    const float w2 = W2[j];
    float contrib[N_GRAPHS];
    for (int g = 0; g < N_GRAPHS; ++g) {
        float acc = b1[j];
        for (int k = 0; k < N_HID; ++k)
            acc += pooled[g * N_HID + k] * W1[k * N_HID + j];
        contrib[g] = fmaxf(acc, 0.0f) * w2;
    }
    for (int g = 0; g < N_GRAPHS; ++g) {
        red[j] = contrib[g];
        __syncthreads();
        for (int s = 64; s > 0; s >>= 1) {
            if (j < s) red[j] += red[j + s];
            __syncthreads();
        }
        if (j == 0) out[g] = red[0] + b2[0];
        __syncthreads();
    }
}

// =====================================================================
// host-side orchestration
// =====================================================================
extern "C" void kernel_launch(void* const* d_in, const int* in_sizes, int n_in,
                              void* d_out, int out_size, void* d_ws, size_t ws_size,
                              hipStream_t stream)
{
    (void)in_sizes; (void)n_in; (void)out_size; (void)ws_size;

    const float* x     = (const float*)d_in[0];
    const int*   e0    = (const int*)  d_in[1];
    const int*   e1    = (const int*)  d_in[2];
    const int*   e2    = (const int*)  d_in[3];
    const int*   batch = (const int*)  d_in[4];
    const float* embW  = (const float*)d_in[5];
    const float* embB  = (const float*)d_in[6];
    const float* convW = (const float*)d_in[7];  // [3][3][128][128]
    const float* rootW = (const float*)d_in[8];  // [3][128][128]
    const float* rootB = (const float*)d_in[9];  // [3][128]
    const float* W1    = (const float*)d_in[10];
    const float* b1    = (const float*)d_in[11];
    const float* W2    = (const float*)d_in[12];
    const float* b2    = (const float*)d_in[13];
    float* out = (float*)d_out;

    // workspace carve-out (256B aligned slabs)
    const size_t per = ((size_t)N_NODES * N_HID * sizeof(float) + 255) & ~(size_t)255;
    char* ws = (char*)d_ws;
    float* h      = (float*)(ws + 0 * per);
    float* o      = (float*)(ws + 1 * per);
    float* m      = (float*)(ws + 2 * per);
    float* agg    = (float*)(ws + 3 * per);
    float* pooled = (float*)(ws + 4 * per);

    const int* srcp[N_REL] = { e0, e1, e2 };   // row 0 = src, row 1 = dst

    const int nElem   = N_NODES * N_HID;                   // 6.4M
    const int n4      = nElem / 4;
    const int strips  = (N_NODES + 31) / 32;               // 1563
    const int eThreads = N_EDGES * 32;
    const int eGrid    = (eThreads + 255) / 256;
    const int v4Grid   = (n4 + 255) / 256;
    const size_t sh64  = (size_t)(IN_FEAT * 136 + 32 * (IN_FEAT + 4)) * sizeof(float);
    const size_t sh128 = (size_t)(N_HID  * 136 + 32 * (N_HID  + 4)) * sizeof(float);

    // 1) node embedding: h = x @ emb_W + emb_b
    gemm_wmma_f32<IN_FEAT><<<strips, 256, sh64, stream>>>(x, embW, embB, h, N_NODES);

    // 2) relational conv layers
    for (int l = 0; l < N_LAYERS; ++l) {
        gemm_wmma_f32<N_HID><<<strips, 256, sh128, stream>>>(
            h, rootW + (size_t)l * N_HID * N_HID, rootB + (size_t)l * N_HID,
            o, N_NODES);
        for (int r = 0; r < N_REL; ++r) {
            gemm_wmma_f32<N_HID><<<strips, 256, sh128, stream>>>(
                h, convW + ((size_t)l * N_REL + r) * N_HID * N_HID, nullptr,
                m, N_NODES);
            fill_u32x4_kernel<<<v4Grid, 256, 0, stream>>>(
                (uint4*)agg, 0xFF800000u /* -inf */, n4);
            scatter_max_kernel<<<eGrid, 256, 0, stream>>>(
                m, srcp[r], srcp[r] + N_EDGES, agg, N_EDGES);
            add_agg_kernel<<<v4Grid, 256, 0, stream>>>(
                (float4*)o, (const float4*)agg, n4);
        }
        relu_copy_kernel<<<v4Grid, 256, 0, stream>>>(
            (float4*)h, (const float4*)o, n4);
    }

    // 3) graph pooling + MLP head
    fill_u32x4_kernel<<<(N_GRAPHS * N_HID / 4 + 255) / 256, 256, 0, stream>>>(
        (uint4*)pooled, 0u, N_GRAPHS * N_HID / 4);
    pool_sum_kernel<<<(N_NODES * 32 + 255) / 256, 256, 0, stream>>>(
        h, batch, pooled, N_NODES);
    mlp_head_kernel<<<1, 128, 0, stream>>>(pooled, W1, b1, W2, b2, out);
}